// DeepAir_1924145348954
// MI455X (gfx1250) — compile-verified
//
#include <hip/hip_runtime.h>
#include <math.h>

#define N_NODES 2000
#define N_EDGES 32000
#define BB 16
#define TT 20
#define OUTF 8
#define HID 24
#define GATES 96      // 4*HID
#define KCAT 32       // OUTF + HID
#define KCATP 33      // padded LDS stride (coprime with 64 banks)
#define FC_OUT 160
#define FC_OUTP 161   // padded
#define GATESP 97     // padded
#define FCC_OUT 20
#define ROWS 320      // B*T

typedef float v2f __attribute__((ext_vector_type(2)));
typedef float v8f __attribute__((ext_vector_type(8)));

__device__ __forceinline__ v8f wmma4_f32(v2f a, v2f b, v8f c) {
  // D = A(16x4) * B(4x16) + C  -> v_wmma_f32_16x16x4_f32
  return __builtin_amdgcn_wmma_f32_16x16x4_f32(false, a, false, b, (short)0, c,
                                               false, false);
}

__device__ __forceinline__ float fast_rcp(float x) {
  return __builtin_amdgcn_rcpf(x);
}
__device__ __forceinline__ float sigm(float x) {
  return fast_rcp(1.0f + __expf(-x));          // v_exp + v_rcp
}
__device__ __forceinline__ float fast_tanh(float x) {
  float t = __expf(2.0f * x);
  return (t - 1.0f) * fast_rcp(t + 1.0f);
}

// ---------------------------------------------------------------------------
// Kernel 1: collapsed edge-GAT. One workgroup per (b,t) graph.
// feat[g][k] = (Wn[k]*S1 + We[k]*S2)/N + gat_b[k],
//   S1 = sum_e a_e*x[src_e], S2 = sum_e a_e*w_e  (edge softmax over dst).
// ---------------------------------------------------------------------------
__global__ __launch_bounds__(256)
void gat_kernel(const float* __restrict__ x, const float* __restrict__ ew,
                const int* __restrict__ src, const int* __restrict__ dst,
                const float* __restrict__ Wn, const float* __restrict__ We,
                const float* __restrict__ al, const float* __restrict__ ae,
                const float* __restrict__ ar, const float* __restrict__ gb,
                float* __restrict__ feats) {
  __shared__ float s_x[N_NODES];      // raw node value
  __shared__ float s_xl[N_NODES];     // x * cL
  __shared__ float s_xr[N_NODES];     // x * cR
  __shared__ int   s_maxi[N_NODES];   // float bits, atomic-max via int/uint trick
  __shared__ float s_den[N_NODES];    // softmax denom, then its reciprocal
  __shared__ float s_r1[8], s_r2[8];
  __shared__ float s_tot[2];

  const int g   = blockIdx.x;
  const int tid = threadIdx.x;
  const float* xg = x  + (size_t)g * N_NODES;
  const float* wg = ew + (size_t)g * N_EDGES;

  float cL = 0.f, cR = 0.f, cE = 0.f;
#pragma unroll
  for (int k = 0; k < OUTF; ++k) {
    float wn = Wn[k];
    cL += wn * al[k];
    cR += wn * ar[k];
    cE += We[k] * ae[k];
  }

  for (int n = tid; n < N_NODES; n += 256) {
    float xv = xg[n];
    s_x[n]    = xv;
    s_xl[n]   = xv * cL;
    s_xr[n]   = xv * cR;
    s_maxi[n] = (int)0xFF800000;   // -inf
    s_den[n]  = 0.f;
  }
  __syncthreads();

  // pass 1: segment max over dst
  for (int e = tid; e < N_EDGES; e += 256) {
    int s = src[e], d = dst[e];
    float sc  = s_xl[s] + fmaf(wg[e], cE, s_xr[d]);
    float evv = sc > 0.f ? sc : 0.2f * sc;        // leaky_relu
    if (evv >= 0.f) atomicMax(&s_maxi[d], __float_as_int(evv));
    else            atomicMin((unsigned int*)&s_maxi[d], __float_as_uint(evv));
  }
  __syncthreads();

  // pass 2: softmax denominator (ds_add_f32)
  for (int e = tid; e < N_EDGES; e += 256) {
    int s = src[e], d = dst[e];
    float sc  = s_xl[s] + fmaf(wg[e], cE, s_xr[d]);
    float evv = sc > 0.f ? sc : 0.2f * sc;
    float num = __expf(evv - __int_as_float(s_maxi[d]));
    atomicAdd(&s_den[d], num);
  }
  __syncthreads();

  // invert denominators once per node (hoists 32000 per-edge divides -> 2000)
  for (int n = tid; n < N_NODES; n += 256) {
    float dn = s_den[n];
    s_den[n] = (dn > 0.f) ? (1.0f / dn) : 0.f;
  }
  __syncthreads();

  // pass 3: weighted edge sums (pure mul now)
  float t1 = 0.f, t2 = 0.f;
  for (int e = tid; e < N_EDGES; e += 256) {
    int s = src[e], d = dst[e];
    float w = wg[e];
    float sc  = s_xl[s] + fmaf(w, cE, s_xr[d]);
    float evv = sc > 0.f ? sc : 0.2f * sc;
    float a = __expf(evv - __int_as_float(s_maxi[d])) * s_den[d];
    t1 = fmaf(a, s_x[s], t1);
    t2 = fmaf(a, w, t2);
  }
  // wave32 reduction then cross-wave
#pragma unroll
  for (int off = 16; off > 0; off >>= 1) {
    t1 += __shfl_down(t1, off, 32);
    t2 += __shfl_down(t2, off, 32);
  }
  if ((tid & 31) == 0) { s_r1[tid >> 5] = t1; s_r2[tid >> 5] = t2; }
  __syncthreads();
  if (tid == 0) {
    float T1 = 0.f, T2 = 0.f;
    for (int i = 0; i < 8; ++i) { T1 += s_r1[i]; T2 += s_r2[i]; }
    s_tot[0] = T1; s_tot[1] = T2;
  }
  __syncthreads();
  if (tid < OUTF) {
    feats[g * OUTF + tid] =
        (Wn[tid] * s_tot[0] + We[tid] * s_tot[1]) * (1.0f / N_NODES) + gb[tid];
  }
}

// ---------------------------------------------------------------------------
// Kernel 2: LSTM via v_wmma_f32_16x16x4_f32.
// A = [feats(8) | h(24)] : 16x32, W = [w_ih | w_hh] : 96x32 (B = W^T tiles).
// 6 waves, one 16-wide gate tile each; 8 K-steps per timestep.
// ---------------------------------------------------------------------------
__global__ __launch_bounds__(192)
void lstm_kernel(const float* __restrict__ feats,
                 const float* __restrict__ w_ih, const float* __restrict__ w_hh,
                 const float* __restrict__ b_ih, const float* __restrict__ b_hh,
                 float* __restrict__ hs) {
  __shared__ float s_w[GATES][KCATP];
  __shared__ float s_a[BB][KCATP];
  __shared__ float s_gates[BB][GATESP];
  __shared__ float s_h[BB][HID];
  __shared__ float s_c[BB][HID];
  __shared__ float s_bias[GATES];

  const int tid  = threadIdx.x;
  const int lane = tid & 31;
  const int wave = tid >> 5;           // 0..5, gate cols [wave*16, wave*16+16)

  for (int i = tid; i < GATES * KCAT; i += 192) {
    int j = i / KCAT, k = i % KCAT;
    s_w[j][k] = (k < OUTF) ? w_ih[j * OUTF + k] : w_hh[j * HID + (k - OUTF)];
  }
  for (int i = tid; i < GATES; i += 192) s_bias[i] = b_ih[i] + b_hh[i];
  for (int i = tid; i < BB * HID; i += 192) {
    (&s_h[0][0])[i] = 0.f;
    (&s_c[0][0])[i] = 0.f;
  }
  __syncthreads();

  const int m     = lane & 15;                 // tile row (batch) / tile col
  const int khalf = (lane < 16) ? 0 : 2;       // K sub-offset per A/B layout
  const int mbase = (lane < 16) ? 0 : 8;       // D rows per C/D layout
  const int j     = wave * 16 + m;             // gate column 0..95

  for (int t = 0; t < TT; ++t) {
    for (int i = tid; i < BB * KCAT; i += 192) {
      int b = i / KCAT, k = i % KCAT;
      s_a[b][k] = (k < OUTF) ? feats[(b * TT + t) * OUTF + k] : s_h[b][k - OUTF];
    }
    __syncthreads();

    v8f acc = {};
#pragma unroll
    for (int kk = 0; kk < KCAT / 4; ++kk) {
      int kb = kk * 4 + khalf;
      v2f a;  a.x  = s_a[m][kb];  a.y  = s_a[m][kb + 1];
      v2f bf; bf.x = s_w[j][kb];  bf.y = s_w[j][kb + 1];
      acc = wmma4_f32(a, bf, acc);
    }
    float bias = s_bias[j];
#pragma unroll
    for (int r = 0; r < 8; ++r) s_gates[mbase + r][j] = acc[r] + bias;
    __syncthreads();

    for (int i = tid; i < BB * HID; i += 192) {
      int b = i / HID, h = i % HID;
      float gi = s_gates[b][h];
      float gf = s_gates[b][HID + h];
      float gg = s_gates[b][2 * HID + h];
      float go = s_gates[b][3 * HID + h];
      float c  = sigm(gf) * s_c[b][h] + sigm(gi) * fast_tanh(gg);
      float hh = sigm(go) * fast_tanh(c);
      s_c[b][h] = c;
      s_h[b][h] = hh;
      hs[(b * TT + t) * HID + h] = hh;
    }
    __syncthreads();
  }
}

// ---------------------------------------------------------------------------
// Kernel 3: fused FC(24->160) + FCC(160->20), WMMA f32 16x16x4.
// One block per 16-row M-tile (20 blocks). Stage-1 intermediate in LDS.
// ---------------------------------------------------------------------------
__global__ __launch_bounds__(320)
void fc_kernel(const float* __restrict__ hs,
               const float* __restrict__ fc_w, const float* __restrict__ fc_b,
               const float* __restrict__ fcc_w, const float* __restrict__ fcc_b,
               float* __restrict__ out) {
  __shared__ float s_o1[BB][FC_OUTP];

  const int tid  = threadIdx.x;
  const int lane = tid & 31;
  const int wave = tid >> 5;                  // 0..9
  const int m0   = blockIdx.x * 16;
  const int n    = lane & 15;
  const int khalf = (lane < 16) ? 0 : 2;
  const int mbase = (lane < 16) ? 0 : 8;

  // stage 1: [16x24] @ fc_w^T[24x160] + fc_b -> LDS
  {
    int jj = wave * 16 + n;                   // 0..159
    v8f acc = {};
#pragma unroll
    for (int kk = 0; kk < HID / 4; ++kk) {
      int kb = kk * 4 + khalf;
      v2f a;  a.x  = hs[(m0 + n) * HID + kb];
              a.y  = hs[(m0 + n) * HID + kb + 1];
      v2f bf; bf.x = fc_w[jj * HID + kb];
              bf.y = fc_w[jj * HID + kb + 1];
      acc = wmma4_f32(a, bf, acc);
    }
    float bias = fc_b[jj];
#pragma unroll
    for (int r = 0; r < 8; ++r) s_o1[mbase + r][jj] = acc[r] + bias;
  }
  __syncthreads();

  // stage 2: [16x160] @ fcc_w^T[160x20] + fcc_b -> d_out (N padded to 32)
  if (wave < 2) {                             // wave-uniform: EXEC all-ones
    int jj = wave * 16 + n;                   // 0..31, valid < 20
    bool valid = jj < FCC_OUT;
    v8f acc = {};
    for (int kk = 0; kk < FC_OUT / 4; ++kk) {
      int kb = kk * 4 + khalf;
      v2f a;  a.x  = s_o1[n][kb];
              a.y  = s_o1[n][kb + 1];
      v2f bf; bf.x = valid ? fcc_w[jj * FC_OUT + kb]     : 0.f;
              bf.y = valid ? fcc_w[jj * FC_OUT + kb + 1] : 0.f;
      acc = wmma4_f32(a, bf, acc);
    }
    if (valid) {
      float bias = fcc_b[jj];
#pragma unroll
      for (int r = 0; r < 8; ++r)
        out[(m0 + mbase + r) * FCC_OUT + jj] = acc[r] + bias;
    }
  }
}

// ---------------------------------------------------------------------------
extern "C" void kernel_launch(void* const* d_in, const int* in_sizes, int n_in,
                              void* d_out, int out_size, void* d_ws, size_t ws_size,
                              hipStream_t stream) {
  const float* x     = (const float*)d_in[0];
  const float* ew    = (const float*)d_in[1];
  const int*   src   = (const int*)  d_in[2];
  const int*   dst   = (const int*)  d_in[3];
  const float* Wn    = (const float*)d_in[4];
  const float* We    = (const float*)d_in[5];
  const float* al    = (const float*)d_in[6];
  const float* ae    = (const float*)d_in[7];
  const float* ar    = (const float*)d_in[8];
  const float* gb    = (const float*)d_in[9];
  const float* w_ih  = (const float*)d_in[10];
  const float* w_hh  = (const float*)d_in[11];
  const float* b_ih  = (const float*)d_in[12];
  const float* b_hh  = (const float*)d_in[13];
  const float* fc_w  = (const float*)d_in[14];
  const float* fc_b  = (const float*)d_in[15];
  const float* fcc_w = (const float*)d_in[16];
  const float* fcc_b = (const float*)d_in[17];

  float* ws    = (float*)d_ws;
  float* feats = ws;              // 320*8  = 2560 floats
  float* hs    = ws + 2560;       // 320*24 = 7680 floats

  gat_kernel<<<ROWS, 256, 0, stream>>>(x, ew, src, dst, Wn, We, al, ae, ar, gb,
                                       feats);
  lstm_kernel<<<1, 192, 0, stream>>>(feats, w_ih, w_hh, b_ih, b_hh, hs);
  fc_kernel<<<ROWS / 16, 320, 0, stream>>>(hs, fc_w, fc_b, fcc_w, fcc_b,
                                           (float*)d_out);
}